// ProteinFeatures_59931973649055
// MI455X (gfx1250) — compile-verified
//
#include <hip/hip_runtime.h>

#define NPE  16
#define NRBF 16
#define FEAT 39   // 16 PE + 16 RBF + 3 dU + 4 quat

// ---------------- fast-math helpers (hardware transcendentals) -------------

__device__ __forceinline__ float fsqrt(float x) { return __builtin_amdgcn_sqrtf(x); }
__device__ __forceinline__ float frcp(float x)  { return __builtin_amdgcn_rcpf(x); }

__device__ __forceinline__ void load_ca(const float* __restrict__ X, int i, float v[3]) {
    // X is (N,4,3); X_ca = X[:,1,:]
    const float* p = X + (size_t)i * 12 + 3;
    v[0] = p[0]; v[1] = p[1]; v[2] = p[2];
}

__device__ __forceinline__ void l2n3(float v[3]) {
    float n = fsqrt(v[0]*v[0] + v[1]*v[1] + v[2]*v[2]);
    float s = frcp(fmaxf(n, 1e-12f));
    v[0] *= s; v[1] *= s; v[2] *= s;
}

__device__ __forceinline__ void cross3(const float a[3], const float b[3], float c[3]) {
    c[0] = a[1]*b[2] - a[2]*b[1];
    c[1] = a[2]*b[0] - a[0]*b[2];
    c[2] = a[0]*b[1] - a[1]*b[0];
}

__device__ __forceinline__ float sgnf(float x) {
    return (x > 0.0f) ? 1.0f : ((x < 0.0f) ? -1.0f : 0.0f);
}

// O-frame for node i (rows: o_1, n_2, o_1 x n_2); zero matrix at chain ends.
__device__ __forceinline__ void compute_O(const float* __restrict__ X, int i, int N, float O[9]) {
    if (i <= 0 || i >= N - 1) {
        #pragma unroll
        for (int k = 0; k < 9; ++k) O[k] = 0.0f;
        return;
    }
    float xm[3], x0[3], xp[3];
    load_ca(X, i - 1, xm); load_ca(X, i, x0); load_ca(X, i + 1, xp);
    float u2[3] = {x0[0]-xm[0], x0[1]-xm[1], x0[2]-xm[2]};
    float u1[3] = {xp[0]-x0[0], xp[1]-x0[1], xp[2]-x0[2]};
    l2n3(u2); l2n3(u1);
    float n2[3]; cross3(u2, u1, n2); l2n3(n2);
    float o1[3] = {u2[0]-u1[0], u2[1]-u1[1], u2[2]-u1[2]};
    l2n3(o1);
    float c2[3]; cross3(o1, n2, c2);
    O[0]=o1[0]; O[1]=o1[1]; O[2]=o1[2];
    O[3]=n2[0]; O[4]=n2[1]; O[5]=n2[2];
    O[6]=c2[0]; O[7]=c2[1]; O[8]=c2[2];
}

// freq_j = 10^(-j/2)  (float32-rounded; matches fp32 exp(2j * -ln(1e4)/16))
__constant__ float c_freq[8] = {
    1.0f, 0.31622776601683794f, 0.1f, 0.031622776601683794f,
    0.01f, 0.0031622776601683794f, 0.001f, 0.00031622776601683794f
};

// sin/cos of fp32 angle a via exact f64 reduction + HW sin (revolutions).
__device__ __forceinline__ void fast_sincos(float a, float& s, float& c) {
    double rev = (double)a * 0.15915494309189535; // a / (2*pi)
    rev -= __builtin_rint(rev);                   // frac in [-0.5, 0.5]
    float rf = (float)rev;
    s = __builtin_amdgcn_sinf(rf);                // sin(2*pi*rf)
    c = __builtin_amdgcn_cosf(rf);                // cos(2*pi*rf)
}

__device__ __forceinline__ float fast_exp_negsq(float t) {
    // exp(-t*t) = 2^(-t*t*log2(e))
    return __builtin_amdgcn_exp2f(t * t * -1.4426950408889634f);
}

// ---------------- main edge-feature kernel ---------------------------------

__global__ __launch_bounds__(256) void edge_feat_kernel(
    const float* __restrict__ X, const int* __restrict__ ei,
    const int* __restrict__ S_id,
    float* __restrict__ outE, float* __restrict__ outEI, float* __restrict__ outLen,
    long long E, int N)
{
    __shared__ int shEI[512];
    __shared__ __align__(16) float shRow[256 * FEAT];   // 39,936 B feature staging

    const int tid = threadIdx.x;
    const long long e = (long long)blockIdx.x * 256 + tid;
    const bool active = (e < E);

    // Stage this block's edge_index slice (rows + cols) into LDS with the
    // CDNA5 async global->LDS path (ASYNCcnt).
    if (active) {
        unsigned lds0 = (unsigned)(size_t)&shEI[tid];
        unsigned lds1 = (unsigned)(size_t)&shEI[256 + tid];
        unsigned off  = (unsigned)(e * 4);
        const int* eiR = ei;
        const int* eiC = ei + E;
        asm volatile("global_load_async_to_lds_b32 %0, %1, %2"
                     :: "v"(lds0), "v"(off), "s"(eiR) : "memory");
        asm volatile("global_load_async_to_lds_b32 %0, %1, %2"
                     :: "v"(lds1), "v"(off), "s"(eiC) : "memory");
        __builtin_prefetch(ei + e + 65536, 0, 0);   // L2 stream-ahead
    }
    asm volatile("s_wait_asynccnt 0x0" ::: "memory");
    __syncthreads();

    if (active) {
        const int r = shEI[tid];
        const int c = shEI[256 + tid];

        float xr[3], xc[3];
        load_ca(X, r, xr);
        load_ca(X, c, xc);

        float dxe[3] = {xc[0]-xr[0], xc[1]-xr[1], xc[2]-xr[2]};
        float len = fsqrt(dxe[0]*dxe[0] + dxe[1]*dxe[1] + dxe[2]*dxe[2]);

        float* row = shRow + tid * FEAT;

        // ---- positional encodings ----
        const float didx = (float)(S_id[c] - S_id[r]);
        #pragma unroll
        for (int j = 0; j < 8; ++j) {
            float a = didx * c_freq[j];
            float s, cc;
            fast_sincos(a, s, cc);
            row[j]     = cc;
            row[8 + j] = s;
        }

        // ---- RBF: mu_k = k*20/15, sigma = 1.25 ----
        #pragma unroll
        for (int k = 0; k < NRBF; ++k) {
            float t = (len - (float)k * (20.0f / 15.0f)) * 0.8f;
            row[NPE + k] = fast_exp_negsq(t);
        }

        // ---- orientation features ----
        float Or[9], Oc[9];
        compute_O(X, r, N, Or);
        compute_O(X, c, N, Oc);

        // dU = l2norm(O[c] @ dXe)
        float dU[3];
        #pragma unroll
        for (int i = 0; i < 3; ++i)
            dU[i] = Oc[i*3+0]*dxe[0] + Oc[i*3+1]*dxe[1] + Oc[i*3+2]*dxe[2];
        l2n3(dU);

        // R = O[r] @ O[c]^T
        float R[9];
        #pragma unroll
        for (int i = 0; i < 3; ++i)
            #pragma unroll
            for (int k = 0; k < 3; ++k)
                R[i*3+k] = Or[i*3+0]*Oc[k*3+0] + Or[i*3+1]*Oc[k*3+1] + Or[i*3+2]*Oc[k*3+2];

        const float Rxx = R[0], Ryy = R[4], Rzz = R[8];
        float m0 = 0.5f * fsqrt(fabsf(1.0f + Rxx - Ryy - Rzz));
        float m1 = 0.5f * fsqrt(fabsf(1.0f - Rxx + Ryy - Rzz));
        float m2 = 0.5f * fsqrt(fabsf(1.0f - Rxx - Ryy + Rzz));
        float q0 = sgnf(R[2*3+1] - R[1*3+2]) * m0;
        float q1 = sgnf(R[0*3+2] - R[2*3+0]) * m1;
        float q2 = sgnf(R[1*3+0] - R[0*3+1]) * m2;
        float q3 = 0.5f * fsqrt(fmaxf(1.0f + Rxx + Ryy + Rzz, 0.0f));
        float qs = frcp(fmaxf(fsqrt(q0*q0 + q1*q1 + q2*q2 + q3*q3), 1e-12f));

        row[32] = dU[0]; row[33] = dU[1]; row[34] = dU[2];
        row[35] = q0*qs; row[36] = q1*qs; row[37] = q2*qs; row[38] = q3*qs;

        // contiguous per-wave outputs: full-cacheline NT stores
        __builtin_nontemporal_store((float)r, outEI + e);
        __builtin_nontemporal_store((float)c, outEI + E + e);
        __builtin_nontemporal_store(len,      outLen + e);
    }
    __syncthreads();

    // ---- stream the dense 256x39 block out of LDS, coalesced b128,
    //      via the CDNA5 async LDS->global store path ----
    const long long e0 = (long long)blockIdx.x * 256;
    const int V = (int)((E - e0 < 256) ? (E - e0) : 256);
    if (V == 256) {
        const size_t baseByte = (size_t)blockIdx.x * (256 * FEAT * 4); // 16B aligned
        #pragma unroll
        for (int i = 0; i < 10; ++i) {
            int idx = i * 256 + tid;             // float4 index, 2496 total
            if (idx < (256 * FEAT) / 4) {
                unsigned ldsA = (unsigned)(size_t)&shRow[idx * 4];
                unsigned goff = (unsigned)(baseByte + (size_t)idx * 16);
                asm volatile("global_store_async_from_lds_b128 %0, %1, %2"
                             :: "v"(goff), "v"(ldsA), "s"(outE) : "memory");
            }
        }
        // ASYNCcnt drains at S_ENDPGM (implicit S_WAIT_IDLE)
    } else if (active) {
        // partial tail block: scalar fallback
        float* rowp = outE + (size_t)e * FEAT;
        const float* row = shRow + tid * FEAT;
        #pragma unroll
        for (int k = 0; k < FEAT; ++k)
            __builtin_nontemporal_store(row[k], rowp + k);
    }
}

// ---------------- ligand (new) edge kernel ----------------------------------

__global__ __launch_bounds__(256) void new_edge_kernel(
    const float* __restrict__ X, const float* __restrict__ PL,
    const int* __restrict__ row_new, const int* __restrict__ col_new,
    float* __restrict__ outEnew, float* __restrict__ outNEI,
    int nr, int N)
{
    int i = blockIdx.x * 256 + threadIdx.x;
    if (i >= nr) return;
    int rn = row_new[i], cn = col_new[i];
    float xr[3]; load_ca(X, rn, xr);
    const float* p = PL + (size_t)cn * 3;
    float d0 = xr[0]-p[0], d1 = xr[1]-p[1], d2 = xr[2]-p[2];
    float len = fsqrt(d0*d0 + d1*d1 + d2*d2);

    float* rowp = outEnew + (size_t)i * FEAT;
    #pragma unroll
    for (int j = 0; j < NPE; ++j) __builtin_nontemporal_store(0.0f, rowp + j);
    #pragma unroll
    for (int k = 0; k < NRBF; ++k) {
        float t = (len - (float)k * (20.0f / 15.0f)) * 0.8f;
        __builtin_nontemporal_store(fast_exp_negsq(t), rowp + NPE + k);
    }
    #pragma unroll
    for (int j = 0; j < 7; ++j) __builtin_nontemporal_store(0.0f, rowp + 32 + j);

    __builtin_nontemporal_store((float)rn,       outNEI + i);
    __builtin_nontemporal_store((float)(cn + N), outNEI + nr + i);
}

// ---------------- launcher ---------------------------------------------------

extern "C" void kernel_launch(void* const* d_in, const int* in_sizes, int n_in,
                              void* d_out, int out_size, void* d_ws, size_t ws_size,
                              hipStream_t stream) {
    const float* X    = (const float*)d_in[0];   // (N,4,3)
    const float* PL   = (const float*)d_in[1];   // (B,3)
    const int*   ei   = (const int*)d_in[2];     // (2,E)
    const int*   S_id = (const int*)d_in[3];     // (N,)
    // d_in[4] = batch (unused by reference outputs)
    const int*   row_new = (const int*)d_in[5];  // (nr,)
    const int*   col_new = (const int*)d_in[6];  // (nr,)

    const long long E  = (long long)(in_sizes[2] / 2);
    const int       N  = in_sizes[3];
    const int       nr = in_sizes[5];

    // d_out layout, flat in return order (all float32):
    //   E(E,39) | edge_index(2,E) | edge_length(E) | new_edge_index(2,nr) | E_new(nr,39)
    float* out     = (float*)d_out;
    float* outE    = out;
    float* outEI   = outE   + (size_t)E * FEAT;
    float* outLen  = outEI  + 2 * (size_t)E;
    float* outNEI  = outLen + (size_t)E;
    float* outEnew = outNEI + 2 * (size_t)nr;

    const int blocksE = (int)((E + 255) / 256);
    edge_feat_kernel<<<blocksE, 256, 0, stream>>>(X, ei, S_id, outE, outEI, outLen, E, N);

    if (nr > 0) {
        const int blocksN = (nr + 255) / 256;
        new_edge_kernel<<<blocksN, 256, 0, stream>>>(X, PL, row_new, col_new, outEnew, outNEI, nr, N);
    }
}